// StateSpaceLayer_60206851555864
// MI455X (gfx1250) — compile-verified
//
#include <hip/hip_runtime.h>
#include <math.h>

// ---------------------------------------------------------------------------
// StateSpaceLayer for MI455X (gfx1250, wave32)
//   BATCH=2, SEQ=64, D_MODEL=256, STATE=64
// Phase 1: h  = gelu(x @ W1 + b1)              (128x256x512, WMMA f32 16x16x4)
// Phase 2: dt = 0.001+0.099*softplus(h@W2+b2)  (128x512x256, WMMA f32 16x16x4)
// Phase 3: SSM scan, one block per channel d. A[d] (16KB) is DMA'd to LDS by
//          the Tensor Data Mover (tensor_load_to_lds + s_wait_tensorcnt),
//          with D# padding (row stride 68 dwords) for conflict-light reads.
//          expm(dt*A)@state applied as an 8-term Taylor matvec (A row in
//          VGPRs, state broadcast via LDS) -- never materializes the 32768
//          dense 64x64 matrix exponentials.
// ---------------------------------------------------------------------------

typedef float v2f __attribute__((ext_vector_type(2)));
typedef float v8f __attribute__((ext_vector_type(8)));
typedef unsigned int v4u __attribute__((ext_vector_type(4)));
typedef int v4i __attribute__((ext_vector_type(4)));
typedef int v8i __attribute__((ext_vector_type(8)));

#define BATCH   2
#define SEQ     64
#define DMODEL  256
#define STATE   64
#define HID     (2 * DMODEL)     // 512
#define MROWS   (BATCH * SEQ)    // 128

#if defined(__has_builtin)
#if __has_builtin(__builtin_amdgcn_tensor_load_to_lds) && \
    __has_builtin(__builtin_amdgcn_s_wait_tensorcnt)
#define HAVE_TDM 1
#endif
#endif
#ifndef HAVE_TDM
#define HAVE_TDM 0
#endif

__device__ __forceinline__ float gelu_exact(float v) {
    return 0.5f * v * (1.0f + erff(v * 0.70710678118654752f));
}

// ---------------------------------------------------------------------------
// GEMM + bias + activation. One wave per block; wave tile = 16 x 64
// (4 fp32 WMMA accumulators, one A-frag amortized over 4 WMMAs).
// K,N compile-time so all strides become immediate offsets.
// ACT=0: GELU, ACT=1: dt = 0.001 + 0.099*softplus(v)
// ---------------------------------------------------------------------------
template <int ACT, int K, int N>
__global__ void __launch_bounds__(32)
gemm_act_kernel(const float* __restrict__ X, const float* __restrict__ W,
                const float* __restrict__ bias, float* __restrict__ Y) {
    const int lane   = threadIdx.x & 31;
    const int col16  = lane & 15;
    const int halfHi = lane >> 4;               // 0: lanes 0-15, 1: lanes 16-31

    const int n0 = blockIdx.x * 64 + col16;     // first of 4 column tiles
    const int m0 = blockIdx.y * 16 + col16;     // A-frag row for this lane

    v8f acc[4];
#pragma unroll
    for (int j = 0; j < 4; ++j) {
        const float bn = bias[n0 + 16 * j];     // fold bias into C
#pragma unroll
        for (int r = 0; r < 8; ++r) acc[j][r] = bn;
    }

    // A 16x4 frag: VGPR0 = K{0|2}, VGPR1 = K{1|3}  (lane-half selects)
    const float* __restrict__ xptr = X + (size_t)m0 * K + (halfHi << 1);
    // B 4x16 frag: VGPR0 = rows {0,1}, VGPR1 = rows {2,3}
    const float* __restrict__ wptr = W + (size_t)halfHi * N + n0;

#pragma unroll 4
    for (int k0 = 0; k0 < K; k0 += 4) {
        v2f a = *(const v2f*)xptr;
        xptr += 4;
        v2f b[4];
#pragma unroll
        for (int j = 0; j < 4; ++j) {
            b[j].x = wptr[16 * j];              // row k0+halfHi
            b[j].y = wptr[2 * N + 16 * j];      // row k0+halfHi+2
        }
        wptr += 4 * N;
#pragma unroll
        for (int j = 0; j < 4; ++j)
            acc[j] = __builtin_amdgcn_wmma_f32_16x16x4_f32(
                false, a, false, b[j], (short)0, acc[j], false, false);
    }

#pragma unroll
    for (int j = 0; j < 4; ++j) {
#pragma unroll
        for (int r = 0; r < 8; ++r) {
            const int m = blockIdx.y * 16 + halfHi * 8 + r;  // D VGPR r = row r/r+8
            float v = acc[j][r];
            if (ACT == 0) {
                v = gelu_exact(v);
            } else {
                const float sp = (v > 20.f) ? v : log1pf(expf(v));
                v = 0.001f + 0.099f * sp;       // DT_MIN + (DT_MAX-DT_MIN)*sp
            }
            Y[(size_t)m * N + n0 + 16 * j] = v;
        }
    }
}

// ---------------------------------------------------------------------------
// SSM scan. One block per channel d; threads 0-63 = batch 0, 64-127 = batch 1.
// Thread i owns state element i and row i of A[d] (64 fp32 in VGPRs).
// ---------------------------------------------------------------------------
#define APAD 68   // dwords per padded A row in LDS (64 data + 4 pad)

__global__ void __launch_bounds__(128)
ssm_scan_kernel(const float* __restrict__ x,  const float* __restrict__ A,
                const float* __restrict__ Bv, const float* __restrict__ Cv,
                const float* __restrict__ Dv, const float* __restrict__ dt,
                float* __restrict__ out) {
    const int d   = blockIdx.x;
    const int tid = threadIdx.x;
    const int b   = tid >> 6;    // batch
    const int i   = tid & 63;    // state index / A row

    __shared__ float sTerm[BATCH][STATE];
    __shared__ float sRed[BATCH][STATE];

    float4 ar[16];               // row i of A[d] in VGPRs

#if HAVE_TDM
    __shared__ float sA[STATE * APAD];           // TDM-padded 64x64 tile
    if (tid < 32) {
        // ---- Tensor DMA descriptor (D#), ISA 8.3/8.4 bit layout ----
        const unsigned long long ga =
            (unsigned long long)(size_t)(A + (size_t)d * STATE * STATE);
        const unsigned ldsOff = (unsigned)(size_t)(void*)sA;  // LDS aperture offset
        v4u g0;
        g0[0] = 1u;                                  // count=1, user mode
        g0[1] = ldsOff;                              // lds_addr (bytes)
        g0[2] = (unsigned)ga;                        // global_addr[31:0]
        g0[3] = ((unsigned)(ga >> 32) & 0x01FFFFFFu) // global_addr[56:32]
                | (2u << 30);                        // type = 2 ("image")
        v8i g1;
        g1[0] = (int)((2u << 16)      // data_size = 4B
                    | (1u << 20)      // pad_enable (LDS row padding)
                    | (5u << 22)      // pad_interval: every 64 dwords
                    | (3u << 25));    // pad_amount: 4 dwords -> stride 68
        g1[1] = (int)(4096u << 16);   // tensor_dim0 = 4096 (1-D view of tile)
        g1[2] = (int)(1u << 16);      // tensor_dim0 hi=0 | tensor_dim1 = 1
        g1[3] = (int)(4096u << 16);   // tensor_dim1 hi=0 | tile_dim0 = 4096
        g1[4] = 1;                    // tile_dim1 = 1, tile_dim2 = 0
        g1[5] = 4096;                 // tensor_dim0_stride[31:0]
        g1[6] = 0;                    // stride hi | tensor_dim1_stride lo
        g1[7] = 0;
        const v4i gz4 = {0, 0, 0, 0};            // groups 2/3 unused (<=2D)
        const v8i gz8 = {0, 0, 0, 0, 0, 0, 0, 0};
        __builtin_amdgcn_tensor_load_to_lds(g0, g1, gz4, gz4, gz8, 0);
        __builtin_amdgcn_s_wait_tensorcnt(0);
    }
    __syncthreads();                  // tile visible to all 4 waves
    {
        const float4* __restrict__ rp = (const float4*)&sA[i * APAD];
#pragma unroll
        for (int q = 0; q < 16; ++q) ar[q] = rp[q];
    }
#else
    {
        const float4* __restrict__ arow =
            (const float4*)(A + (((size_t)d * STATE) + i) * STATE);
#pragma unroll
        for (int q = 0; q < 16; ++q) ar[q] = arow[q];
    }
#endif

    const float Bi = Bv[(size_t)d * STATE + i];
    const float Ci = Cv[(size_t)d * STATE + i];
    const float Dd = Dv[d];

    const float kInv[8] = {1.f, 0.5f, 1.f / 3.f, 0.25f,
                           0.2f, 1.f / 6.f, 1.f / 7.f, 0.125f};

    float s = 0.f;
    for (int t = 0; t < SEQ; ++t) {
        const int idx = (b * SEQ + t) * DMODEL + d;
        const float dtv = dt[idx];
        const float xv  = x[idx];

        // --- z = expm(dtv * A) @ s, Taylor order 8 (||dt*A|| ~ 0.2) ---
        sTerm[b][i] = s;
        float accv = s;
        __syncthreads();
#pragma unroll
        for (int k = 0; k < 8; ++k) {
            float mv = 0.f;
#pragma unroll
            for (int q = 0; q < 16; ++q) {   // A-row in VGPRs, state broadcast
                mv = fmaf(ar[q].x, sTerm[b][4 * q + 0], mv);
                mv = fmaf(ar[q].y, sTerm[b][4 * q + 1], mv);
                mv = fmaf(ar[q].z, sTerm[b][4 * q + 2], mv);
                mv = fmaf(ar[q].w, sTerm[b][4 * q + 3], mv);
            }
            __syncthreads();                 // all reads of old term done
            const float nterm = mv * dtv * kInv[k];
            sTerm[b][i] = nterm;
            accv += nterm;
            __syncthreads();                 // new term visible
        }

        // --- state update + output projection ---
        s = accv + dtv * xv * Bi;            // + Bd * x_t
        sRed[b][i] = Ci * s;                 // y_t = C . s
        __syncthreads();
#pragma unroll
        for (int off = 32; off >= 1; off >>= 1) {
            if (i < off) sRed[b][i] += sRed[b][i + off];
            __syncthreads();
        }
        if (i == 0) out[idx] = sRed[b][0] + Dd * xv;
        // sRed is not rewritten until >=2 barriers into step t+1: safe.
    }
}

// ---------------------------------------------------------------------------
extern "C" void kernel_launch(void* const* d_in, const int* in_sizes, int n_in,
                              void* d_out, int out_size, void* d_ws, size_t ws_size,
                              hipStream_t stream) {
    (void)in_sizes; (void)n_in; (void)out_size; (void)ws_size;
    const float* x  = (const float*)d_in[0];  // (2,64,256)
    const float* A  = (const float*)d_in[1];  // (256,64,64)
    const float* Bv = (const float*)d_in[2];  // (256,64,1)
    const float* Cv = (const float*)d_in[3];  // (256,1,64)
    const float* Dv = (const float*)d_in[4];  // (256,)
    const float* W1 = (const float*)d_in[5];  // (256,512)
    const float* b1 = (const float*)d_in[6];  // (512,)
    const float* W2 = (const float*)d_in[7];  // (512,256)
    const float* b2 = (const float*)d_in[8];  // (256,)
    float* out = (float*)d_out;               // (2,64,256)

    float* h   = (float*)d_ws;                // 128x512 scratch
    float* dtb = h + (size_t)MROWS * HID;     // 128x256 scratch

    // h = gelu(x @ W1 + b1): wave tile 16x64, grid (512/64, 128/16)
    gemm_act_kernel<0, DMODEL, HID><<<dim3(HID / 64, MROWS / 16), dim3(32), 0, stream>>>(
        x, W1, b1, h);
    // dt = 0.001 + 0.099*softplus(h @ W2 + b2): grid (256/64, 128/16)
    gemm_act_kernel<1, HID, DMODEL><<<dim3(DMODEL / 64, MROWS / 16), dim3(32), 0, stream>>>(
        h, W2, b2, dtb);
    // scan: one block per channel
    ssm_scan_kernel<<<dim3(DMODEL), dim3(128), 0, stream>>>(
        x, A, Bv, Cv, Dv, dtb, out);
}